// GruModel_58617713656043
// MI455X (gfx1250) — compile-verified
//
#include <hip/hip_runtime.h>
#include <math.h>

#define VOCAB  5000
#define HIDDEN 512
#define BATCH  64
#define SEQ    256
#define G3     (3 * HIDDEN)   // 1536

typedef __attribute__((ext_vector_type(2))) float v2f;
typedef __attribute__((ext_vector_type(8))) float v8f;

// ---------------------------------------------------------------------------
// WMMA f32 16x16x4 fragment layouts (CDNA5 ISA 7.12.2)
// A 16x4 (MxK): lanes 0-15 -> M=lane, {K=k,k+1}; lanes 16-31 -> M=lane-16, {K=k+2,k+3}
// B 4x16 (KxN): lanes 0-15 -> N=lane, {K=k,k+1}; lanes 16-31 -> N=lane-16, {K=k+2,k+3}
// C/D 16x16:    VGPR r, lanes 0-15 -> M=r, N=lane; lanes 16-31 -> M=r+8, N=lane-16
// ---------------------------------------------------------------------------

__device__ __forceinline__ v8f wmma_f32(v2f a, v2f b, v8f c) {
    return __builtin_amdgcn_wmma_f32_16x16x4_f32(
        /*neg_a=*/false, a, /*neg_b=*/false, b,
        /*c_mod=*/(short)0, c, /*reuse_a=*/false, /*reuse_b=*/false);
}

// One wave computes a 16x64 strip of C = A (MxK, row-major) @ B^T
// (B supplied row-major as Brow[n][k], ldb = K-stride).
// NGUARD: N not a multiple of the strip (clamp B addresses, guard stores).
// Garbage columns from clamped addresses only affect D columns >= nmax,
// which are never stored -> no value masking, inner loop is branchless.
template <bool NGUARD>
__device__ __forceinline__ void wave_gemm_16x64(
    const float* __restrict__ A, int lda,
    const float* __restrict__ Brow, int ldb,
    float* __restrict__ C, int ldc,
    int m0, int n0, int K, int nmax,
    const float* __restrict__ bias, int lane) {

    const int ln = lane & 15;
    const int kh = (lane >> 4) << 1;           // +0 (lanes 0-15) or +2 (lanes 16-31)

    const float* ap = A + (long)(m0 + ln) * lda + kh;

    const float* bp[4];
#pragma unroll
    for (int j = 0; j < 4; ++j) {
        int col = n0 + j * 16 + ln;
        if (NGUARD) col = (col < nmax) ? col : (nmax - 1);   // address clamp only
        bp[j] = Brow + (long)col * ldb + kh;
    }

    v8f c0, c1, c2, c3;
#pragma unroll
    for (int r = 0; r < 8; ++r) { c0[r] = 0.f; c1[r] = 0.f; c2[r] = 0.f; c3[r] = 0.f; }

#pragma unroll 8
    for (int k = 0; k < K; k += 4) {
        v2f a  = *(const v2f*)(ap + k);
        v2f b0 = *(const v2f*)(bp[0] + k);
        v2f b1 = *(const v2f*)(bp[1] + k);
        v2f b2 = *(const v2f*)(bp[2] + k);
        v2f b3 = *(const v2f*)(bp[3] + k);
        c0 = wmma_f32(a, b0, c0);
        c1 = wmma_f32(a, b1, c1);
        c2 = wmma_f32(a, b2, c2);
        c3 = wmma_f32(a, b3, c3);
    }

    const int rbase = m0 + ((lane >> 4) << 3); // +0 or +8
    v8f acc[4] = {c0, c1, c2, c3};
#pragma unroll
    for (int j = 0; j < 4; ++j) {
        int col = n0 + j * 16 + ln;
        if (!NGUARD || col < nmax) {
            float bv = bias ? bias[col] : 0.0f;
#pragma unroll
            for (int r = 0; r < 8; ++r)
                C[(long)(rbase + r) * ldc + col] = acc[j][r] + bv;
        }
    }
}

// ---------------------------------------------------------------------------
// Per-step recurrent GEMM: gh[b][g] = sum_k h[b][k] * W_hh[g][k]
// M=64 (4 tiles) x N=1536, K=512. 96 waves -> 12 blocks x 256 threads.
// ---------------------------------------------------------------------------
__global__ void gru_hh_gemm(const float* __restrict__ h,
                            const float* __restrict__ W_hh,
                            float* __restrict__ gh) {
    int wave = blockIdx.x * (blockDim.x >> 5) + (threadIdx.x >> 5);
    int lane = threadIdx.x & 31;
    int m0 = (wave & 3) * 16;       // 4 M-tiles
    int n0 = (wave >> 2) * 64;      // 24 N-groups

    wave_gemm_16x64<false>(h, HIDDEN, W_hh, HIDDEN, gh, G3,
                           m0, n0, HIDDEN, G3, nullptr, lane);
}

// ---------------------------------------------------------------------------
// Per-step gating with the one-hot input projection fused as a gather from
// W_ih (L2-resident, 30 MB < 192 MB L2). One thread per (b, j).
// ---------------------------------------------------------------------------
__global__ void gru_gate(const float* __restrict__ gh,
                         const int*   __restrict__ X,
                         const float* __restrict__ W_ih,
                         const float* __restrict__ b_ih,
                         const float* __restrict__ b_hh,
                         const float* __restrict__ h_prev,
                         float* __restrict__ h_out,
                         int s) {
    int idx = blockIdx.x * blockDim.x + threadIdx.x;
    if (idx >= BATCH * HIDDEN) return;
    int b = idx >> 9;               // / HIDDEN
    int j = idx & (HIDDEN - 1);

    int token = X[b * SEQ + s];

    float gi_r = W_ih[(long)j * VOCAB + token]                + b_ih[j];
    float gi_z = W_ih[(long)(HIDDEN + j) * VOCAB + token]     + b_ih[HIDDEN + j];
    float gi_n = W_ih[(long)(2 * HIDDEN + j) * VOCAB + token] + b_ih[2 * HIDDEN + j];

    const float* ghb = gh + (long)b * G3;
    float gh_r = ghb[j]              + b_hh[j];
    float gh_z = ghb[HIDDEN + j]     + b_hh[HIDDEN + j];
    float gh_n = ghb[2 * HIDDEN + j] + b_hh[2 * HIDDEN + j];

    float r = 1.0f / (1.0f + __expf(-(gi_r + gh_r)));
    float z = 1.0f / (1.0f + __expf(-(gi_z + gh_z)));
    float n = tanhf(gi_n + r * gh_n);

    float hp = h_prev[idx];
    h_out[idx] = (1.0f - z) * n + z * hp;
}

// ---------------------------------------------------------------------------
// Output GEMM: out[m][n] = sum_k Y[m][k] * W_out[n][k] + b_out[n]
// M=16384 (1024 tiles), N=5000 (tail-guarded), K=512.
// 79 N-groups of 64 -> 1024*79 = 80896 waves = 10112 blocks x 8 waves.
// ---------------------------------------------------------------------------
#define OUT_NGRP 79

__global__ void out_gemm(const float* __restrict__ Y,
                         const float* __restrict__ W_out,
                         const float* __restrict__ b_out,
                         float* __restrict__ out) {
    int wave = blockIdx.x * (blockDim.x >> 5) + (threadIdx.x >> 5);
    int lane = threadIdx.x & 31;
    int mt = wave / OUT_NGRP;
    int ng = wave % OUT_NGRP;
    if (mt >= (SEQ * BATCH) / 16) return;   // uniform per wave

    wave_gemm_16x64<true>(Y, HIDDEN, W_out, HIDDEN, out, VOCAB,
                          mt * 16, ng * 64, HIDDEN, VOCAB, b_out, lane);
}

// Copy final hidden state hT = Y[S-1] into the tail of d_out.
__global__ void copy_hT(const float* __restrict__ src, float* __restrict__ dst) {
    int i = blockIdx.x * blockDim.x + threadIdx.x;
    if (i < BATCH * HIDDEN) dst[i] = src[i];
}

// ---------------------------------------------------------------------------
extern "C" void kernel_launch(void* const* d_in, const int* in_sizes, int n_in,
                              void* d_out, int out_size, void* d_ws, size_t ws_size,
                              hipStream_t stream) {
    const int*   X     = (const int*)  d_in[0];
    const float* state = (const float*)d_in[1];   // (1, B, H)
    const float* W_ih  = (const float*)d_in[2];   // (3H, V)
    const float* W_hh  = (const float*)d_in[3];   // (3H, H)
    const float* b_ih  = (const float*)d_in[4];
    const float* b_hh  = (const float*)d_in[5];
    const float* W_out = (const float*)d_in[6];   // (V, H)
    const float* b_out = (const float*)d_in[7];
    float* out = (float*)d_out;                   // (S*B, V) then (B, H)

    // workspace layout
    float* gh = (float*)d_ws;                     // (B, 3H)
    float* Y  = gh + (long)BATCH * G3;            // (S, B, H)

    for (int s = 0; s < SEQ; ++s) {
        const float* h_prev = (s == 0) ? state : (Y + (long)(s - 1) * BATCH * HIDDEN);
        float*       h_out  = Y + (long)s * BATCH * HIDDEN;

        gru_hh_gemm<<<12, 256, 0, stream>>>(h_prev, W_hh, gh);
        gru_gate<<<(BATCH * HIDDEN) / 256, 256, 0, stream>>>(
            gh, X, W_ih, b_ih, b_hh, h_prev, h_out, s);
    }

    out_gemm<<<10112, 256, 0, stream>>>(Y, W_out, b_out, out);

    copy_hT<<<(BATCH * HIDDEN) / 256, 256, 0, stream>>>(
        Y + (long)(SEQ - 1) * BATCH * HIDDEN, out + (long)SEQ * BATCH * VOCAB);
}